// MinSumRelOut_25374666784832
// MI455X (gfx1250) — compile-verified
//
#include <hip/hip_runtime.h>
#include <math.h>

#define B_DIM 512
#define I_DIM 256
#define O_DIM 512

typedef __attribute__((ext_vector_type(2))) float v2f;
typedef __attribute__((ext_vector_type(8))) float v8f;

// ---------------------------------------------------------------------------
// Kernel 1a: reciprocal + transpose.
//   UT[i][b] = 1 / x[b][i]        (LDS-tiled 32x32, coalesced both sides)
// Hoists every division out of the WMMA inner loop (each 1/x element was
// being recomputed 32x there) and lays A out so each lane's K-pair is one
// contiguous 8-byte load.
// ---------------------------------------------------------------------------
#define TS 32
__global__ void msro_recip_transpose_kernel(const float* __restrict__ x,
                                            float* __restrict__ ut) {
    __shared__ float tile[TS][TS + 1];
    const int bx = blockIdx.x % (I_DIM / TS);   // i-tile (8)
    const int by = blockIdx.x / (I_DIM / TS);   // b-tile (16)
    const int i0 = bx * TS, b0 = by * TS;
    const int tx = threadIdx.x & 31;
    const int ty = threadIdx.x >> 5;            // 32x8 threads
#pragma unroll
    for (int j = 0; j < TS; j += 8)             // coalesced read of x
        tile[ty + j][tx] = 1.0f / x[(b0 + ty + j) * I_DIM + i0 + tx];
    __syncthreads();
#pragma unroll
    for (int j = 0; j < TS; j += 8)             // coalesced write of UT
        ut[(i0 + ty + j) * B_DIM + b0 + tx] = tile[tx][ty + j];
}

// ---------------------------------------------------------------------------
// Kernel 1b: column reductions.
//   tid in [0, I):      Sx[i]    = sum_b (1 - x[b,i])
//   tid in [I, I+O):    sinv[o]  = sum_i 1 / w[i,o]
//                       scomp[o] = sum_i (1 - w[i,o])
// ---------------------------------------------------------------------------
__global__ void msro_sums_kernel(const float* __restrict__ x,
                                 const float* __restrict__ w,
                                 float* __restrict__ Sx,
                                 float* __restrict__ sinv,
                                 float* __restrict__ scomp) {
    int tid = blockIdx.x * blockDim.x + threadIdx.x;
    if (tid < I_DIM) {
        float s = 0.0f;
        for (int b = 0; b < B_DIM; ++b) s += 1.0f - x[b * I_DIM + tid];
        Sx[tid] = s;
    } else if (tid < I_DIM + O_DIM) {
        int o = tid - I_DIM;
        float si = 0.0f, sc = 0.0f;
        for (int i = 0; i < I_DIM; ++i) {
            float wv = w[i * O_DIM + o];
            si += 1.0f / wv;
            sc += 1.0f - wv;
        }
        sinv[o]  = si;
        scomp[o] = sc;
    }
}

// ---------------------------------------------------------------------------
// Kernel 2: rel_x via WMMA f32 GEMM.
//   G[i,o] = sum_b UT[i,b] * t[b,o]   ->   rel_x[i,o] = 1 - (B - G)/Sx[i]
// One wave32 per 16(i) x 32(o) tile: A fragment (b64 load) feeds two
// V_WMMA_F32_16X16X4_F32s.  Inner loop is now pure loads + WMMA.
// A layout (32-bit 16x4): lanes 0-15 M=0..15 / K=0,1; lanes 16-31 K=2,3.
// C/D: vgpr v, half h, lane r -> (M = v + 8h, N = r).  EXEC all-ones.
// ---------------------------------------------------------------------------
__global__ void msro_relx_wmma_kernel(const float* __restrict__ ut,
                                      const float* __restrict__ t,
                                      const float* __restrict__ Sx,
                                      float* __restrict__ relx) {
    const int lane = threadIdx.x & 31;
    const int wave = (blockIdx.x * blockDim.x + threadIdx.x) >> 5;  // 0..255
    const int ti  = wave >> 4;          // 16 i-tiles
    const int to2 = wave & 15;          // 16 o-pair tiles
    const int i0 = ti * 16;
    const int oA = to2 * 32, oB = oA + 16;
    const int half = lane >> 4;
    const int r    = lane & 15;

    const float* ap  = ut + (i0 + r) * B_DIM + 2 * half;  // K-pair contiguous
    const float* tpA = t + oA + r;
    const float* tpB = t + oB + r;

    v8f acc0 = {}, acc1 = {};
#pragma unroll 8
    for (int kb = 0; kb < B_DIM; kb += 4) {
        const v2f a = *(const v2f*)(ap + kb);              // global_load_b64
        const int b0 = kb + 2 * half;
        v2f bA, bB;
        bA[0] = tpA[(b0)     * O_DIM];
        bA[1] = tpA[(b0 + 1) * O_DIM];
        bB[0] = tpB[(b0)     * O_DIM];
        bB[1] = tpB[(b0 + 1) * O_DIM];
        acc0 = __builtin_amdgcn_wmma_f32_16x16x4_f32(
            false, a, false, bA, (short)0, acc0, false, false);
        acc1 = __builtin_amdgcn_wmma_f32_16x16x4_f32(
            false, a, false, bB, (short)0, acc1, false, false);
    }

#pragma unroll
    for (int v = 0; v < 8; ++v) {
        const int i = i0 + v + 8 * half;
        const float sx = Sx[i];
        relx[i * O_DIM + oA + r] = 1.0f - ((float)B_DIM - acc0[v]) / sx;
        relx[i * O_DIM + oB + r] = 1.0f - ((float)B_DIM - acc1[v]) / sx;
    }
}

// ---------------------------------------------------------------------------
// Kernel 3: dual argmin over I + gathered outputs.
//   ind_x[b,o] = argmin_i ( x[b,i] + rel_x[i,o] - x[b,i]*rel_x[i,o] )
//   rel_w[b,o] = 1 - (I - t[b,o]*sinv[o]) / scomp[o]   (fully factorized)
//   ind_w[b,o] = argmin_i ( w[i,o] + rel_w[b,o] - w[i,o]*rel_w[b,o] )
//   out_x = max(x[b,ind_x], w[ind_x,o]);  out_w likewise with ind_w.
// Block = 16b x 16o tile; i staged through LDS in chunks of 64; x tile
// padded to keep the column walk conflict-free.  Strict '<' + ascending i
// == first-occurrence argmin (jnp semantics).
// ---------------------------------------------------------------------------
#define ICHUNK 64
__global__ void msro_argmin_out_kernel(const float* __restrict__ x,
                                       const float* __restrict__ w,
                                       const float* __restrict__ t,
                                       const float* __restrict__ relx,
                                       const float* __restrict__ sinv,
                                       const float* __restrict__ scomp,
                                       float* __restrict__ out_x,
                                       float* __restrict__ out_w) {
    __shared__ float sxx[16][ICHUNK + 1];
    __shared__ float sw [ICHUNK][16];
    __shared__ float sr [ICHUNK][16];

    const int tid = threadIdx.x;
    const int tx = tid & 15;                // o within tile
    const int ty = tid >> 4;                // b within tile
    const int b0 = (blockIdx.x >> 5) * 16;  // 32 b-tiles
    const int o0 = (blockIdx.x & 31) * 16;  // 32 o-tiles
    const int b = b0 + ty;
    const int o = o0 + tx;

    const float tv = t[b * O_DIM + o];
    const float rw = 1.0f - ((float)I_DIM - tv * sinv[o]) / scomp[o];

    float fmin = INFINITY, gmin = INFINITY;
    int ixm = 0, iwm = 0;

    for (int ic = 0; ic < I_DIM; ic += ICHUNK) {
#pragma unroll
        for (int j = 0; j < 4; ++j) {
            const int idx = tid + j * 256;
            const int br = idx >> 6, icc = idx & 63;   // 16 x 64 x-tile
            sxx[br][icc] = x[(b0 + br) * I_DIM + ic + icc];
            const int ir = idx >> 4, oc = idx & 15;    // 64 x 16 w / relx
            sw[ir][oc] = w[(ic + ir) * O_DIM + o0 + oc];
            sr[ir][oc] = relx[(ic + ir) * O_DIM + o0 + oc];
        }
        __syncthreads();

#pragma unroll 4
        for (int k = 0; k < ICHUNK; ++k) {
            const float xi = sxx[ty][k];
            const float rv = sr[k][tx];
            const float f = xi + rv - xi * rv;
            if (f < fmin) { fmin = f; ixm = ic + k; }
            const float wi = sw[k][tx];
            const float g = wi + rw - wi * rw;
            if (g < gmin) { gmin = g; iwm = ic + k; }
        }
        __syncthreads();
    }

    out_x[b * O_DIM + o] = fmaxf(x[b * I_DIM + ixm], w[ixm * O_DIM + o]);
    out_w[b * O_DIM + o] = fmaxf(x[b * I_DIM + iwm], w[iwm * O_DIM + o]);
}

// ---------------------------------------------------------------------------
// Launch: {recip-transpose, sums} -> relx(WMMA) -> argmin/out, one stream.
// Workspace: rel_x (I*O) + UT (B*I) + Sx (I) + sinv (O) + scomp (O) ~ 1.03 MB.
// ---------------------------------------------------------------------------
extern "C" void kernel_launch(void* const* d_in, const int* in_sizes, int n_in,
                              void* d_out, int out_size, void* d_ws, size_t ws_size,
                              hipStream_t stream) {
    const float* x = (const float*)d_in[0];   // (B, I)
    const float* w = (const float*)d_in[1];   // (I, O)
    const float* t = (const float*)d_in[2];   // (B, O)
    float* out = (float*)d_out;               // chosen_x (B,O) ++ chosen_w (B,O)

    float* ws    = (float*)d_ws;
    float* relx  = ws;                         // I*O
    float* ut    = relx + I_DIM * O_DIM;       // B*I   (transposed reciprocal)
    float* Sx    = ut + B_DIM * I_DIM;         // I
    float* sinv  = Sx + I_DIM;                 // O
    float* scomp = sinv + O_DIM;               // O

    // 1a) UT = (1/x)^T : 16x8 = 128 tiles of 32x32
    msro_recip_transpose_kernel<<<128, 256, 0, stream>>>(x, ut);
    // 1b) column sums: I + O = 768 threads
    msro_sums_kernel<<<3, 256, 0, stream>>>(x, w, Sx, sinv, scomp);
    // 2) rel_x via WMMA: 256 waves (16x32-wide tiles), 8 waves/block
    msro_relx_wmma_kernel<<<32, 256, 0, stream>>>(ut, t, Sx, relx);
    // 3) argmins + outputs: 32x32 tiles of (B,O)
    msro_argmin_out_kernel<<<1024, 256, 0, stream>>>(
        x, w, t, relx, sinv, scomp, out, out + B_DIM * O_DIM);
}